// TransformerEncoder_4741643895126
// MI455X (gfx1250) — compile-verified
//
#include <hip/hip_runtime.h>
#include <math.h>

typedef __attribute__((ext_vector_type(2))) float v2f;
typedef __attribute__((ext_vector_type(8))) float v8f;

#define Bb 4
#define Ss 1024
#define Dd 1024
#define Hh 16
#define HDd 64
#define RDd 4096
#define NROW (Bb * Ss)      // 4096 token rows

__device__ __forceinline__ float gelu_erf(float x) {
    // exact GELU: 0.5*x*(1+erf(x/sqrt(2))) matching torch default
    return 0.5f * x * (1.0f + erff(x * 0.70710678118654752440f));
}

// ---------------------------------------------------------------------------
// CDNA5 async global->LDS DMA (ASYNCcnt-tracked), inline asm per ISA 08.
// The low 32 bits of a flat pointer into __shared__ are the wave-relative
// LDS offset, which is what the instruction's VDST operand expects.
// ---------------------------------------------------------------------------
__device__ __forceinline__ void async_b128(void* lds, const float* g) {
    asm volatile("global_load_async_to_lds_b128 %0, %1, off"
                 :: "v"((unsigned)(uintptr_t)lds), "v"(g)
                 : "memory");
}
#define WAIT_ASYNC(n) asm volatile("s_wait_asynccnt %0" :: "n"(n) : "memory")

// ---------------------------------------------------------------------------
// Fused LayerNorm: one block per row (D=1024), 256 threads, 1 float4/thread.
// ---------------------------------------------------------------------------
__global__ __launch_bounds__(256) void ln_kernel(const float* __restrict__ x,
                                                 const float* __restrict__ g,
                                                 const float* __restrict__ b,
                                                 float* __restrict__ out) {
    __shared__ float red0[8];
    __shared__ float red1[8];
    const int row  = blockIdx.x;
    const int t    = threadIdx.x;
    const int lane = t & 31;
    const int wid  = t >> 5;

    float4 v = ((const float4*)(x + (size_t)row * Dd))[t];
    float s  = v.x + v.y + v.z + v.w;
    float ss = v.x * v.x + v.y * v.y + v.z * v.z + v.w * v.w;
#pragma unroll
    for (int off = 16; off > 0; off >>= 1) {
        s  += __shfl_down(s, off, 32);
        ss += __shfl_down(ss, off, 32);
    }
    if (lane == 0) { red0[wid] = s; red1[wid] = ss; }
    __syncthreads();
    if (t == 0) {
        float a0 = 0.f, a1 = 0.f;
#pragma unroll
        for (int i = 0; i < 8; ++i) { a0 += red0[i]; a1 += red1[i]; }
        red0[0] = a0; red1[0] = a1;
    }
    __syncthreads();
    const float mean = red0[0] * (1.0f / Dd);
    const float var  = red1[0] * (1.0f / Dd) - mean * mean;
    const float rstd = rsqrtf(var + 1e-5f);

    float4 gg = ((const float4*)g)[t];
    float4 bb = ((const float4*)b)[t];
    float4 o;
    o.x = (v.x - mean) * rstd * gg.x + bb.x;
    o.y = (v.y - mean) * rstd * gg.y + bb.y;
    o.z = (v.z - mean) * rstd * gg.z + bb.z;
    o.w = (v.w - mean) * rstd * gg.w + bb.w;
    ((float4*)(out + (size_t)row * Dd))[t] = o;
}

// ---------------------------------------------------------------------------
// WMMA fp32 GEMM: C[M,N] = epilogue(A[M,K] * B[K,N] + bias[N]) (+ residual)
// 256 threads (8 waves). Block tile 128x128, wave tile 32x64, K-tile 32,
// double-buffered LDS filled by async global->LDS DMA.
// MODE: 0 = bias, 1 = bias+GELU, 2 = bias+residual.
// ---------------------------------------------------------------------------
template <int MODE>
__global__ __launch_bounds__(256) void gemm_kernel(const float* __restrict__ A,
                                                   const float* __restrict__ Bm,
                                                   const float* __restrict__ bias,
                                                   const float* __restrict__ Res,
                                                   float* __restrict__ C,
                                                   int M, int N, int K) {
    // stride 36 floats (144B): 16B-aligned rows; A-frag reads hit banks
    // {4k}+c (lanes 0-15) vs {4k}+c+2 (lanes 16-31) -> conflict-free.
    // stride 144 floats (576B): halves of a wave offset by 32 banks.
    __shared__ __align__(16) float As[2][128][36];
    __shared__ __align__(16) float Bs[2][32][144];

    const int t    = threadIdx.x;
    const int lane = t & 31;
    const int wid  = t >> 5;
    const int wr   = wid >> 1;          // 0..3 : wave row (32 rows each)
    const int wc   = wid & 1;           // 0..1 : wave col (64 cols each)
    const int bm   = blockIdx.y * 128;
    const int bn   = blockIdx.x * 128;
    const int col  = lane & 15;
    const int kb   = (lane >> 4) << 1;  // K-pair select for fragments

    // Issue one K-tile (128x32 of A, 32x128 of B) as 8 async b128 DMAs/thread.
    auto issue_tile = [&](int kt, int buf) {
#pragma unroll
        for (int i = 0; i < 4; ++i) {
            int s = t + i * 256;                 // 0..1023
            int row = s >> 3, c4 = (s & 7) << 2;
            async_b128(&As[buf][row][c4],
                       A + (size_t)(bm + row) * K + kt + c4);
        }
#pragma unroll
        for (int i = 0; i < 4; ++i) {
            int s = t + i * 256;
            int row = s >> 5, c4 = (s & 31) << 2;
            async_b128(&Bs[buf][row][c4],
                       Bm + (size_t)(kt + row) * N + bn + c4);
        }
    };

    v8f acc[2][4] = {};
    const int NT = K >> 5;              // K / 32 tiles
    issue_tile(0, 0);

    for (int it = 0; it < NT; ++it) {
        const int buf = it & 1;
        if (it + 1 < NT) {
            issue_tile((it + 1) << 5, buf ^ 1);
            WAIT_ASYNC(8);              // tile `it`'s 8 DMAs retired
        } else {
            WAIT_ASYNC(0);
        }
        __syncthreads();                // tile data visible to all waves

#pragma unroll
        for (int k0 = 0; k0 < 32; k0 += 4) {
            v2f a[2], bf[4];
#pragma unroll
            for (int mi = 0; mi < 2; ++mi) {
                a[mi].x = As[buf][wr * 32 + mi * 16 + col][k0 + kb];
                a[mi].y = As[buf][wr * 32 + mi * 16 + col][k0 + kb + 1];
            }
#pragma unroll
            for (int ni = 0; ni < 4; ++ni) {
                bf[ni].x = Bs[buf][k0 + kb][wc * 64 + ni * 16 + col];
                bf[ni].y = Bs[buf][k0 + kb + 1][wc * 64 + ni * 16 + col];
            }
#pragma unroll
            for (int mi = 0; mi < 2; ++mi)
#pragma unroll
                for (int ni = 0; ni < 4; ++ni)
                    acc[mi][ni] = __builtin_amdgcn_wmma_f32_16x16x4_f32(
                        false, a[mi], false, bf[ni], (short)0, acc[mi][ni],
                        false, false);
        }
        __syncthreads();                // all reads done before buf is refilled
    }

    // epilogue: C/D layout -> (row = slot + 8*(lane>=16), col = lane&15)
    const int rhalf = (lane >> 4) << 3;
#pragma unroll
    for (int mi = 0; mi < 2; ++mi) {
#pragma unroll
        for (int ni = 0; ni < 4; ++ni) {
#pragma unroll
            for (int i = 0; i < 8; ++i) {
                int r = bm + wr * 32 + mi * 16 + i + rhalf;
                int c = bn + wc * 64 + ni * 16 + col;
                float val = acc[mi][ni][i] + bias[c];
                if (MODE == 1) val = gelu_erf(val);
                if (MODE == 2) val += Res[(size_t)r * N + c];
                C[(size_t)r * N + c] = val;
            }
        }
    }
}

// ---------------------------------------------------------------------------
// Flash attention (fp32 WMMA): one block = (b, h, 64-query tile), 4 waves,
// each wave owns 16 query rows. K/V streamed in 16-key tiles through
// double-buffered LDS filled by async DMA.
// ---------------------------------------------------------------------------
__global__ __launch_bounds__(128) void attn_kernel(const float* __restrict__ Q,
                                                   const float* __restrict__ Kt,
                                                   const float* __restrict__ V,
                                                   float* __restrict__ O) {
    __shared__ __align__(16) float Qs[64][68];       // 272B rows, 16B aligned
    __shared__ __align__(16) float Ks[2][16][68];
    __shared__ __align__(16) float Vs[2][16][80];    // halves 32 banks apart
    __shared__ float Ps[4][16][17];                  // per-wave P tile

    const int qb = blockIdx.x;      // S/64 tiles
    const int h  = blockIdx.y;
    const int b  = blockIdx.z;
    const int t    = threadIdx.x;
    const int lane = t & 31;
    const int w    = t >> 5;
    const int col  = lane & 15;
    const int kb   = (lane >> 4) << 1;
    const int rhalf = (lane >> 4) << 3;
    const size_t base = (size_t)b * Ss * Dd + (size_t)h * HDd;

    // Q tile 64x64 = 1024 float4, 8 per thread (stays resident)
#pragma unroll
    for (int i = 0; i < 8; ++i) {
        int s   = t + i * 128;
        int row = s >> 4, c4 = (s & 15) << 2;
        float4 val = *(const float4*)(Q + base + (size_t)(qb * 64 + row) * Dd + c4);
        *(float4*)(&Qs[row][c4]) = val;
    }

    auto issue_kv = [&](int kt, int buf) {           // 4 async DMAs / thread
#pragma unroll
        for (int i = 0; i < 2; ++i) {
            int s   = t + i * 128;
            int row = s >> 4, c4 = (s & 15) << 2;
            async_b128(&Ks[buf][row][c4],
                       Kt + base + (size_t)(kt + row) * Dd + c4);
            async_b128(&Vs[buf][row][c4],
                       V + base + (size_t)(kt + row) * Dd + c4);
        }
    };

    float m_i[8], l_i[8];
#pragma unroll
    for (int i = 0; i < 8; ++i) { m_i[i] = -INFINITY; l_i[i] = 0.f; }
    v8f acc[4] = {};

    const int NT = Ss >> 4;             // 64 key tiles
    issue_kv(0, 0);

    for (int it = 0; it < NT; ++it) {
        const int buf = it & 1;
        if (it + 1 < NT) {
            issue_kv((it + 1) << 4, buf ^ 1);
            WAIT_ASYNC(4);
        } else {
            WAIT_ASYNC(0);
        }
        __syncthreads();                // K/V tile ready (covers Qs on it==0)

        // S = Q * K^T   (16 queries x 16 keys, K-dim = 64)
        v8f s8 = {};
#pragma unroll
        for (int k0 = 0; k0 < HDd; k0 += 4) {
            v2f a, bf;
            a.x  = Qs[w * 16 + col][k0 + kb];
            a.y  = Qs[w * 16 + col][k0 + kb + 1];
            bf.x = Ks[buf][col][k0 + kb];   // B[kdim][key] = K[key][kdim]
            bf.y = Ks[buf][col][k0 + kb + 1];
            s8 = __builtin_amdgcn_wmma_f32_16x16x4_f32(false, a, false, bf,
                                                       (short)0, s8, false, false);
        }

        // online softmax update (row stats via 16-lane half reductions)
        float alpha[8];
#pragma unroll
        for (int i = 0; i < 8; ++i) {
            float sv = s8[i] * 0.125f;      // 1/sqrt(64)
            float mx = sv;
#pragma unroll
            for (int off = 8; off > 0; off >>= 1)
                mx = fmaxf(mx, __shfl_xor(mx, off, 32));
            float mnew = fmaxf(m_i[i], mx);
            float p    = __expf(sv - mnew);
            float sum  = p;
#pragma unroll
            for (int off = 8; off > 0; off >>= 1)
                sum += __shfl_xor(sum, off, 32);
            alpha[i] = __expf(m_i[i] - mnew);
            l_i[i]   = l_i[i] * alpha[i] + sum;
            m_i[i]   = mnew;
            Ps[w][i + rhalf][col] = p;      // C-layout -> row-major LDS
        }
#pragma unroll
        for (int ni = 0; ni < 4; ++ni)
#pragma unroll
            for (int i = 0; i < 8; ++i) acc[ni][i] *= alpha[i];
        __syncthreads();                // P visible before fragment reads

        // ctx += P * V   (A = P 16x16, B = V 16x64)
#pragma unroll
        for (int k0 = 0; k0 < 16; k0 += 4) {
            v2f a;
            a.x = Ps[w][col][k0 + kb];
            a.y = Ps[w][col][k0 + kb + 1];
#pragma unroll
            for (int ni = 0; ni < 4; ++ni) {
                v2f bf;
                bf.x = Vs[buf][k0 + kb][ni * 16 + col];
                bf.y = Vs[buf][k0 + kb + 1][ni * 16 + col];
                acc[ni] = __builtin_amdgcn_wmma_f32_16x16x4_f32(
                    false, a, false, bf, (short)0, acc[ni], false, false);
            }
        }
        __syncthreads();                // buf fully read before next-iter DMA
    }

    // normalize and store ctx
#pragma unroll
    for (int ni = 0; ni < 4; ++ni) {
#pragma unroll
        for (int i = 0; i < 8; ++i) {
            int row = qb * 64 + w * 16 + i + rhalf;
            int c   = ni * 16 + col;
            O[base + (size_t)row * Dd + c] = acc[ni][i] / l_i[i];
        }
    }
}

// ---------------------------------------------------------------------------
extern "C" void kernel_launch(void* const* d_in, const int* in_sizes, int n_in,
                              void* d_out, int out_size, void* d_ws, size_t ws_size,
                              hipStream_t stream) {
    (void)in_sizes; (void)n_in; (void)out_size; (void)ws_size;
    const float* x     = (const float*)d_in[0];
    const float* ln1_g = (const float*)d_in[1];
    const float* ln1_b = (const float*)d_in[2];
    const float* wq    = (const float*)d_in[3];
    const float* bq    = (const float*)d_in[4];
    const float* wk    = (const float*)d_in[5];
    const float* bk    = (const float*)d_in[6];
    const float* wv    = (const float*)d_in[7];
    const float* bv    = (const float*)d_in[8];
    const float* wo    = (const float*)d_in[9];
    const float* bo    = (const float*)d_in[10];
    const float* ln2_g = (const float*)d_in[11];
    const float* ln2_b = (const float*)d_in[12];
    const float* w1    = (const float*)d_in[13];
    const float* b1    = (const float*)d_in[14];
    const float* w2    = (const float*)d_in[15];
    const float* b2    = (const float*)d_in[16];

    const size_t TOK = (size_t)NROW * Dd;   // 4M floats
    float* ws  = (float*)d_ws;
    float* h   = ws;             // ln1 output          [4M]
    float* q   = ws + 1 * TOK;   //                     [4M]
    float* k   = ws + 2 * TOK;   //                     [4M]
    float* v   = ws + 3 * TOK;   //                     [4M]
    float* x2  = ws + 4 * TOK;   // after attn residual [4M]
    float* a1  = ws + 5 * TOK;   // MLP hidden          [16M]
    float* ctx = h;              // h dead after QKV
    float* h2  = q;              // q dead after attention

    dim3 gQKV(Dd / 128, NROW / 128);      // (8, 32)
    dim3 gMLP1(RDd / 128, NROW / 128);    // (32, 32)
    dim3 gMLP2(Dd / 128, NROW / 128);     // (8, 32)

    ln_kernel<<<NROW, 256, 0, stream>>>(x, ln1_g, ln1_b, h);
    gemm_kernel<0><<<gQKV, 256, 0, stream>>>(h, wq, bq, nullptr, q, NROW, Dd, Dd);
    gemm_kernel<0><<<gQKV, 256, 0, stream>>>(h, wk, bk, nullptr, k, NROW, Dd, Dd);
    gemm_kernel<0><<<gQKV, 256, 0, stream>>>(h, wv, bv, nullptr, v, NROW, Dd, Dd);
    attn_kernel<<<dim3(Ss / 64, Hh, Bb), 128, 0, stream>>>(q, k, v, ctx);
    gemm_kernel<2><<<gQKV, 256, 0, stream>>>(ctx, wo, bo, x, x2, NROW, Dd, Dd);
    ln_kernel<<<NROW, 256, 0, stream>>>(x2, ln2_g, ln2_b, h2);
    gemm_kernel<1><<<gMLP1, 256, 0, stream>>>(h2, w1, b1, nullptr, a1, NROW, RDd, Dd);
    gemm_kernel<2><<<gMLP2, 256, 0, stream>>>(a1, w2, b2, x2, (float*)d_out,
                                              NROW, Dd, RDd);
}